// TANN_9251359556272
// MI455X (gfx1250) — compile-verified
//
#include <hip/hip_runtime.h>
#include <stdint.h>

// H[i,j] = max_k (|W[i,k]| + |W[j,k]|)   (max-plus Gram matrix)
// N = 2048, K = 10000, f32 in / f32 out.
//
// Strategy: 128x128 output tiles over the upper triangle (136 block-pairs),
// split-K by 2 (-> 272 blocks for occupancy on a large-WGP-count part),
// async global->LDS double-buffered K-streaming (ASYNCcnt), 8x8 f32
// micro-tiles, abs folded into VALU src modifiers, max folded to v_max3.
// K-halves merged with global_atomic_max_u32 on raw f32 bits (monotone for
// the nonnegative values here -> bit-exact, order-independent).

#define N_DIM 2048
#define K_DIM 10000
#define KC 40                        // k-chunk; 250 * 40 == 10000 exactly
#define PITCH 44                     // LDS row pitch in floats (40 data + 4 pad -> conflict-free b128)
#define TILE_ROWS 128
#define TILE_FP (TILE_ROWS * PITCH)  // 5632 floats per tile
#define TILE_BYTES (TILE_FP * 4)     // 22528 bytes
#define NCHUNK (K_DIM / KC)          // 250 total chunks
#define KSPLIT 2
#define NCHUNK_PER (NCHUNK / KSPLIT) // 125 chunks per K-half
#define SEGS_PER_ROW (KC * 4 / 16)   // 10 x 16B segments per row-chunk
#define SEGS_PER_THREAD (TILE_ROWS * SEGS_PER_ROW / 256)  // 5

// CDNA5 async copy: copies 16 bytes from the global address g to the LDS address lds_off, tracked by ASYNCcnt.
__device__ __forceinline__ void async_copy16(uint32_t lds_off, const float* g) {
  asm volatile("global_load_async_to_lds_b128 %0, %1, off"
               :: "v"(lds_off), "v"(g) : "memory");
}

__global__ void zero_fill_kernel(uint4* __restrict__ p) {
  p[(size_t)blockIdx.x * blockDim.x + threadIdx.x] = uint4{0u, 0u, 0u, 0u};
}

__global__ __launch_bounds__(256) void tropical_gram_kernel(const float* __restrict__ W,
                                                            unsigned* __restrict__ H) {
  // A0 | B0 | A1 | B1  (double-buffered i-tile / j-tile) = 88 KB
  __shared__ alignas(16) float smem[4 * TILE_FP];

  const int tid = threadIdx.x;
  const int tx = tid & 15;   // j micro-index
  const int ty = tid >> 4;   // i micro-index

  // Decode upper-triangular block pair (bi <= bj) from linear block id (136 pairs).
  int rem = blockIdx.x;
  int bi = 0;
  while (rem >= (16 - bi)) { rem -= (16 - bi); ++bi; }
  const int bj = bi + rem;
  const int I0 = bi * TILE_ROWS;
  const int J0 = bj * TILE_ROWS;

  // K-half handled by this block.
  const int chunk0 = (int)blockIdx.y * NCHUNK_PER;
  const int kbase = chunk0 * KC;

  // Per-thread loader assignment: 5 x 16B segments per tile, uniform across all lanes.
  uint32_t ldsA[SEGS_PER_THREAD];
  const float* gA[SEGS_PER_THREAD];
  const float* gB[SEGS_PER_THREAD];
  const uint32_t smem_base = (uint32_t)(uintptr_t)(&smem[0]);
#pragma unroll
  for (int r = 0; r < SEGS_PER_THREAD; ++r) {
    const int s = tid + 256 * r;
    const int row = s / SEGS_PER_ROW;
    const int seg = s % SEGS_PER_ROW;
    ldsA[r] = smem_base + (uint32_t)(row * (PITCH * 4) + seg * 16);
    gA[r] = W + (size_t)(I0 + row) * K_DIM + kbase + seg * 4;
    gB[r] = W + (size_t)(J0 + row) * K_DIM + kbase + seg * 4;
  }

  float acc[8][8];
#pragma unroll
  for (int u = 0; u < 8; ++u)
#pragma unroll
    for (int v = 0; v < 8; ++v) acc[u][v] = 0.0f;  // |a|+|b| >= 0

  // Prologue: first chunk of this K-half -> buffer 0 (10 async instructions/thread).
#pragma unroll
  for (int r = 0; r < SEGS_PER_THREAD; ++r) {
    async_copy16(ldsA[r], gA[r]);
    async_copy16(ldsA[r] + TILE_BYTES, gB[r]);
  }

  for (int c = 0; c < NCHUNK_PER; ++c) {
    const int cur = c & 1;
    if (c + 1 < NCHUNK_PER) {
      const int koff = (c + 1) * KC;
      const uint32_t boff = (uint32_t)((1 - cur) * 2 * TILE_BYTES);
#pragma unroll
      for (int r = 0; r < SEGS_PER_THREAD; ++r) {
        async_copy16(ldsA[r] + boff, gA[r] + koff);
        async_copy16(ldsA[r] + boff + TILE_BYTES, gB[r] + koff);
      }
      // 10 newly issued still in flight; anything older (== chunk c) must be done.
      asm volatile("s_wait_asynccnt 10" ::: "memory");
    } else {
      asm volatile("s_wait_asynccnt 0" ::: "memory");
    }
    __syncthreads();

    const float4* As = (const float4*)(smem + cur * 2 * TILE_FP);
    const float4* Bs = (const float4*)(smem + cur * 2 * TILE_FP + TILE_FP);
#pragma unroll 2
    for (int kk = 0; kk < KC / 4; ++kk) {
      float4 av[8], bv[8];
#pragma unroll
      for (int u = 0; u < 8; ++u) av[u] = As[(ty + 16 * u) * (PITCH / 4) + kk];
#pragma unroll
      for (int v = 0; v < 8; ++v) bv[v] = Bs[(tx + 16 * v) * (PITCH / 4) + kk];
#pragma unroll
      for (int u = 0; u < 8; ++u)
#pragma unroll
        for (int v = 0; v < 8; ++v) {
          // fabs folds into VALU src modifiers; fmax(fmax(x,a),b) -> v_max3
          const float s0 = __builtin_fabsf(av[u].x) + __builtin_fabsf(bv[v].x);
          const float s1 = __builtin_fabsf(av[u].y) + __builtin_fabsf(bv[v].y);
          const float s2 = __builtin_fabsf(av[u].z) + __builtin_fabsf(bv[v].z);
          const float s3 = __builtin_fabsf(av[u].w) + __builtin_fabsf(bv[v].w);
          acc[u][v] = fmaxf(fmaxf(acc[u][v], s0), s1);
          acc[u][v] = fmaxf(fmaxf(acc[u][v], s2), s3);
        }
    }
    __syncthreads();
  }

  // Merge K-halves with integer atomic max on raw bits (values >= 0 -> monotone,
  // bit-exact, order-independent). Also write the mirrored tile.
#pragma unroll
  for (int u = 0; u < 8; ++u) {
    const int i = I0 + ty + 16 * u;
#pragma unroll
    for (int v = 0; v < 8; ++v) {
      const int j = J0 + tx + 16 * v;
      const unsigned bits = __float_as_uint(acc[u][v]);
      atomicMax(&H[(size_t)i * N_DIM + j], bits);
      atomicMax(&H[(size_t)j * N_DIM + i], bits);
    }
  }
}

extern "C" void kernel_launch(void* const* d_in, const int* in_sizes, int n_in,
                              void* d_out, int out_size, void* d_ws, size_t ws_size,
                              hipStream_t stream) {
  (void)in_sizes; (void)n_in; (void)d_ws; (void)ws_size; (void)out_size;
  const float* W = (const float*)d_in[0];

  // Zero-fill output so the atomic-max accumulation is idempotent and
  // independent of prior d_out contents (harness doesn't re-poison).
  const int fill_blocks = (N_DIM * N_DIM) / (256 * 4);  // uint4 per thread
  zero_fill_kernel<<<fill_blocks, 256, 0, stream>>>((uint4*)d_out);

  const int npairs = (16 * 17) / 2;  // 136 upper-triangular 128x128 block pairs
  tropical_gram_kernel<<<dim3(npairs, KSPLIT), 256, 0, stream>>>(W, (unsigned*)d_out);
}